// GNNModel_74174085202016
// MI455X (gfx1250) — compile-verified
//
#include <hip/hip_runtime.h>
#include <cstdint>
#include <cstddef>

typedef __attribute__((ext_vector_type(2))) float v2f;
typedef __attribute__((ext_vector_type(8))) float v8f;

// ---------------------------------------------------------------------------
// Degree / normalization
// ---------------------------------------------------------------------------
__global__ void k_init_deg(float* __restrict__ deg, int n) {
  int i = blockIdx.x * blockDim.x + threadIdx.x;
  if (i < n) deg[i] = 1.0f;  // self-loop contributes 1 to every node's degree
}

__global__ void k_count_deg(const long long* __restrict__ dst,
                            float* __restrict__ deg, int e) {
  int i = blockIdx.x * blockDim.x + threadIdx.x;
  if (i < e) atomicAdd(deg + (int)dst[i], 1.0f);
}

__global__ void k_rsqrt_inplace(float* __restrict__ d, int n) {
  int i = blockIdx.x * blockDim.x + threadIdx.x;
  if (i < n) d[i] = rsqrtf(d[i]);  // deg >= 1 always (self-loops)
}

// count must be a multiple of 4 (all our regions are 32n/64n floats)
__global__ void k_zero4(float4* __restrict__ p, int count4) {
  int i = blockIdx.x * blockDim.x + threadIdx.x;
  if (i < count4) p[i] = make_float4(0.0f, 0.0f, 0.0f, 0.0f);
}

// ---------------------------------------------------------------------------
// Layer 1 dense transform: h1[n][j] = sum_k x[n][k] * W1[k][j]   (5 -> 64)
// K=5 is too skinny for WMMA; VALU is fine (64 MFLOP total).
// ---------------------------------------------------------------------------
__global__ void k_h1(const float* __restrict__ x, const float* __restrict__ W1,
                     float* __restrict__ h1, int n) {
  int gid = blockIdx.x * blockDim.x + threadIdx.x;
  int node = gid >> 6;
  int j = gid & 63;
  if (node >= n) return;
  const float* xr = x + (size_t)node * 5;
  float acc = 0.0f;
#pragma unroll
  for (int k = 0; k < 5; ++k) acc = fmaf(xr[k], W1[k * 64 + j], acc);
  h1[gid] = acc;
}

// ---------------------------------------------------------------------------
// Edge scatter: acc[dst] += h[src] * dinv[src]*dinv[dst]
// One wave (32 lanes) per edge. FEAT=64: float2 per lane; FEAT=32: 1 per lane.
// Gathers/atomics land in L2 (h fits easily in the 192MB L2).
// ---------------------------------------------------------------------------
template <int FEAT>
__global__ void k_scatter(const long long* __restrict__ src,
                          const long long* __restrict__ dst,
                          const float* __restrict__ dinv,
                          const float* __restrict__ h,
                          float* __restrict__ acc, int e) {
  long long gid = (long long)blockIdx.x * blockDim.x + threadIdx.x;
  long long edge = gid >> 5;
  if (edge >= e) return;
  int lane = (int)(gid & 31);
  int s = (int)src[edge];
  int d = (int)dst[edge];
  float w = dinv[s] * dinv[d];
  if (FEAT == 64) {
    v2f v = *(const v2f*)(h + (size_t)s * 64 + 2 * lane);
    atomicAdd(acc + (size_t)d * 64 + 2 * lane + 0, v.x * w);
    atomicAdd(acc + (size_t)d * 64 + 2 * lane + 1, v.y * w);
  } else {
    float v = h[(size_t)s * 32 + lane];
    atomicAdd(acc + (size_t)d * 32 + lane, v * w);
  }
}

// ---------------------------------------------------------------------------
// Post-aggregation (layer 1): x1 = relu(a1 + selfloop(h1*dinv^2) + b1)
// ---------------------------------------------------------------------------
__global__ void k_postagg1(const float* __restrict__ a1,
                           const float* __restrict__ h1,
                           const float* __restrict__ dinv,
                           const float* __restrict__ b1,
                           float* __restrict__ x1, int n) {
  int gid = blockIdx.x * blockDim.x + threadIdx.x;
  int node = gid >> 6;
  int j = gid & 63;
  if (node >= n) return;
  float di = dinv[node];
  float v = a1[gid] + h1[gid] * di * di + b1[j];
  x1[gid] = fmaxf(v, 0.0f);
}

// ---------------------------------------------------------------------------
// Layer 2 dense transform via full-fp32 WMMA: h2 = x1 @ W2  (64 -> 32)
// One wave computes a 16x16 output tile; K=64 as 16 x V_WMMA_F32_16X16X4_F32.
//
// All 16 A-fragments + 16 B-fragments are loaded up front (single load
// clause, one wait) and the 16 WMMAs then issue back-to-back as a pure
// C-accumulation chain. OOB rows are handled by clamping the node index
// (loads duplicate row n-1; stores are guarded), so EXEC stays all-ones
// with no per-iteration exec save/restore.
//
// VGPR layouts per CDNA5 ISA 7.12.2:
//   A 16x4 f32 (2 VGPRs): lanes 0-15 row M=lane hold K={k0,k0+1};
//                         lanes 16-31 row M=lane-16 hold K={k0+2,k0+3}.
//   B 4x16 f32 (2 VGPRs): mirrored K split, N striped across lanes 0-15.
//   C/D 16x16 f32 (8 VGPRs): vgpr v -> M = v + 8*(lane>=16), N = lane&15.
// ---------------------------------------------------------------------------
__global__ void k_h2_wmma(const float* __restrict__ x1,
                          const float* __restrict__ W2,
                          float* __restrict__ h2, int n) {
  int wavesPerBlock = blockDim.x >> 5;
  int tile = blockIdx.x * wavesPerBlock + ((int)threadIdx.x >> 5);
  int mtiles = (n + 15) >> 4;
  if (tile >= mtiles * 2) return;  // uniform per wave: whole wave exits together

  int lane  = threadIdx.x & 31;
  int row   = lane & 15;   // M (for A) / N (for B and C/D)
  int khalf = lane >> 4;   // which K pair this lane holds
  int mbase = (tile >> 1) << 4;
  int nbase = (tile & 1) << 4;
  int node  = mbase + row;
  if (node >= n) node = n - 1;  // clamp: garbage rows computed but never stored
  int col = nbase + row;

  const float* arow = x1 + (size_t)node * 64 + 2 * khalf;
  const float* bcol = W2 + (size_t)2 * khalf * 32 + col;

  v2f af[16];
  v2f bf[16];
#pragma unroll
  for (int kk = 0; kk < 16; ++kk) {
    af[kk] = *(const v2f*)(arow + 4 * kk);
    bf[kk].x = bcol[(size_t)(4 * kk + 0) * 32];
    bf[kk].y = bcol[(size_t)(4 * kk + 1) * 32];
  }

  v8f acc = {};
#pragma unroll
  for (int kk = 0; kk < 16; ++kk) {
    // 8 args: (neg_a, A, neg_b, B, c_mod, C, reuse_a, reuse_b)
    acc = __builtin_amdgcn_wmma_f32_16x16x4_f32(
        false, af[kk], false, bf[kk], (short)0, acc, false, false);
  }

#pragma unroll
  for (int v = 0; v < 8; ++v) {
    int m = mbase + v + 8 * khalf;
    if (m < n) h2[(size_t)m * 32 + nbase + row] = acc[v];
  }
}

// ---------------------------------------------------------------------------
// Final fused kernel: layer-2 post-agg (self-loop + b2 + relu) then the
// 32->16->2 MLP head, one node per thread, weights staged in LDS.
// ---------------------------------------------------------------------------
__global__ void k_final(const float* __restrict__ a2,
                        const float* __restrict__ h2,
                        const float* __restrict__ dinv,
                        const float* __restrict__ b2,
                        const float* __restrict__ fcW1,
                        const float* __restrict__ fcb1,
                        const float* __restrict__ fcW2,
                        const float* __restrict__ fcb2,
                        float* __restrict__ out, int n) {
  __shared__ float sW1[32 * 16];
  __shared__ float sb1[16];
  __shared__ float sW2[16 * 2];
  __shared__ float sb2[2];
  __shared__ float sB2[32];

  for (int i = threadIdx.x; i < 512; i += blockDim.x) sW1[i] = fcW1[i];
  if (threadIdx.x < 32) {
    sW2[threadIdx.x] = fcW2[threadIdx.x];
    sB2[threadIdx.x] = b2[threadIdx.x];
  }
  if (threadIdx.x < 16) sb1[threadIdx.x] = fcb1[threadIdx.x];
  if (threadIdx.x < 2)  sb2[threadIdx.x] = fcb2[threadIdx.x];
  __syncthreads();

  int node = blockIdx.x * blockDim.x + threadIdx.x;
  if (node >= n) return;

  float di = dinv[node];
  float d2 = di * di;
  float x2[32];
#pragma unroll
  for (int j = 0; j < 32; ++j) {
    float v = a2[(size_t)node * 32 + j] + h2[(size_t)node * 32 + j] * d2 + sB2[j];
    x2[j] = fmaxf(v, 0.0f);
  }
  float y[16];
#pragma unroll
  for (int m = 0; m < 16; ++m) {
    float acc = sb1[m];
#pragma unroll
    for (int j = 0; j < 32; ++j) acc = fmaf(x2[j], sW1[j * 16 + m], acc);
    y[m] = fmaxf(acc, 0.0f);
  }
#pragma unroll
  for (int t = 0; t < 2; ++t) {
    float acc = sb2[t];
#pragma unroll
    for (int m = 0; m < 16; ++m) acc = fmaf(y[m], sW2[m * 2 + t], acc);
    out[(size_t)node * 2 + t] = acc;
  }
}

// ---------------------------------------------------------------------------
// Host launcher
// ---------------------------------------------------------------------------
extern "C" void kernel_launch(void* const* d_in, const int* in_sizes, int n_in,
                              void* d_out, int out_size, void* d_ws, size_t ws_size,
                              hipStream_t stream) {
  const float*     x    = (const float*)d_in[0];      // [n, 5]
  const long long* ei   = (const long long*)d_in[1];  // [2, e] int64
  const float*     W1   = (const float*)d_in[2];      // [5, 64]
  const float*     b1   = (const float*)d_in[3];      // [64]
  const float*     W2   = (const float*)d_in[4];      // [64, 32]
  const float*     b2   = (const float*)d_in[5];      // [32]
  const float*     fcW1 = (const float*)d_in[6];      // [32, 16]
  const float*     fcb1 = (const float*)d_in[7];      // [16]
  const float*     fcW2 = (const float*)d_in[8];      // [16, 2]
  const float*     fcb2 = (const float*)d_in[9];      // [2]
  float*           out  = (float*)d_out;              // [n, 2]

  int n = in_sizes[0] / 5;
  int e = in_sizes[1] / 2;
  const long long* src = ei;
  const long long* dst = ei + e;

  // Workspace layout (floats): dinv[n] | bufA[64n] | bufB[64n]  (~51.6 MB)
  float* ws   = (float*)d_ws;
  float* dinv = ws;
  float* bufA = ws + n;
  float* bufB = bufA + (size_t)64 * n;
  float* h1 = bufA;                    // [n,64]
  float* a1 = bufB;                    // [n,64] aggregation target
  float* x1 = bufB;                    // relu'd in place over a1
  float* h2 = bufA;                    // [n,32] reuses bufA after h1 consumed
  float* a2 = bufA + (size_t)32 * n;   // [n,32] upper half of bufA

  const int B = 256;
  auto cdiv = [](long long a, long long b) { return (unsigned)((a + b - 1) / b); };

  // 1) degree (with self-loop) -> dinv = rsqrt(deg)
  k_init_deg<<<cdiv(n, B), B, 0, stream>>>(dinv, n);
  k_count_deg<<<cdiv(e, B), B, 0, stream>>>(dst, dinv, e);
  k_rsqrt_inplace<<<cdiv(n, B), B, 0, stream>>>(dinv, n);

  // 2) layer 1: transform, scatter, self-loop + bias + relu
  k_zero4<<<cdiv((long long)16 * n, B), B, 0, stream>>>((float4*)a1, 16 * n);
  k_h1<<<cdiv((long long)64 * n, B), B, 0, stream>>>(x, W1, h1, n);
  k_scatter<64><<<cdiv((long long)e * 32, B), B, 0, stream>>>(src, dst, dinv, h1, a1, e);
  k_postagg1<<<cdiv((long long)64 * n, B), B, 0, stream>>>(a1, h1, dinv, b1, x1, n);

  // 3) layer 2: WMMA transform (h1 fully consumed above; bufA reusable)
  k_zero4<<<cdiv((long long)8 * n, B), B, 0, stream>>>((float4*)a2, 8 * n);
  {
    int tiles = ((n + 15) >> 4) * 2;       // 16-row tiles x 2 column tiles
    int wavesPerBlock = B / 32;
    k_h2_wmma<<<cdiv(tiles, wavesPerBlock), B, 0, stream>>>(x1, W2, h2, n);
  }
  k_scatter<32><<<cdiv((long long)e * 32, B), B, 0, stream>>>(src, dst, dinv, h2, a2, e);

  // 4) layer-2 post-agg + MLP head, fused
  k_final<<<cdiv(n, B), B, 0, stream>>>(a2, h2, dinv, b2, fcW1, fcb1, fcW2, fcb2, out, n);
}